// RiemannFormerAttention_5686536700144
// MI455X (gfx1250) — compile-verified
//
#include <hip/hip_runtime.h>
#include <hip/hip_bf16.h>
#include <math.h>

typedef __attribute__((ext_vector_type(2))) float v2f;
typedef __attribute__((ext_vector_type(8))) float v8f;
typedef __attribute__((ext_vector_type(4))) unsigned int u32x4;
typedef __attribute__((ext_vector_type(8))) int i32x8;
typedef __attribute__((ext_vector_type(4))) int i32x4;

#if __has_builtin(__builtin_amdgcn_tensor_load_to_lds) && \
    __has_builtin(__builtin_amdgcn_s_wait_tensorcnt)
#define USE_TDM 1
#else
#define USE_TDM 0
#endif

// D = A(16x4 f32) * B(4x16 f32) + C(16x16 f32)  -> v_wmma_f32_16x16x4_f32
__device__ __forceinline__ v8f wmma4(v2f a, v2f b, v8f c) {
    return __builtin_amdgcn_wmma_f32_16x16x4_f32(
        /*neg_a=*/false, a, /*neg_b=*/false, b,
        /*c_mod=*/(short)0, c, /*reuse_a=*/false, /*reuse_b=*/false);
}

#if USE_TDM
// Tensor Data Mover: DMA a (rows x cols) fp32 tile (global row stride `gstride`
// elements) into LDS at byte offset lds_off, packed row-major. D# fields per
// CDNA5 ISA ch.8: group0 = count/lds_addr/global_addr/type, group1 =
// data_size + tensor dims + tile dims + dim0 stride. Groups 2/3 zero (<=2D).
// 6-arg builtin form (clang-23 / therock-10.0 toolchain).
__device__ __forceinline__ void tdm_load_2d(unsigned lds_off, const float* g,
                                            unsigned cols, unsigned rows,
                                            unsigned gstride) {
    const unsigned long long ga = (unsigned long long)(size_t)g;
    u32x4 g0;
    g0.x = 1u;                                            // count=1 (valid user D#)
    g0.y = lds_off;                                       // lds_addr (bytes)
    g0.z = (unsigned)(ga & 0xffffffffu);                  // global_addr[31:0]
    g0.w = (unsigned)((ga >> 32) & 0x01ffffffu) | (2u << 30); // [56:32] | type=2
    i32x8 g1;
    g1[0] = (int)(2u << 16);                              // data_size=2 (4B), mask=0
    g1[1] = (int)((cols & 0xffffu) << 16);                // tensor_dim0[15:0]
    g1[2] = (int)(((cols >> 16) & 0xffffu) | ((rows & 0xffffu) << 16)); // d0 hi | d1 lo
    g1[3] = (int)(((rows >> 16) & 0xffffu) | ((cols & 0xffffu) << 16)); // d1 hi | tile_dim0
    g1[4] = (int)(rows & 0xffffu);                        // tile_dim1 | tile_dim2=0
    g1[5] = (int)gstride;                                 // tensor_dim0_stride[31:0]
    g1[6] = 0;                                            // stride0 hi | stride1 lo
    g1[7] = 0;
    const i32x4 z4 = {0, 0, 0, 0};
    const i32x8 z8 = {0, 0, 0, 0, 0, 0, 0, 0};
    __builtin_amdgcn_tensor_load_to_lds(g0, g1, z4, z4, z8, 0);
}
#endif

// Issue a contiguous 16KB (4096-float) tile copy global -> LDS.
__device__ __forceinline__ void tile16k_issue(float* dst, const float* src) {
#if USE_TDM
    if (threadIdx.x < 32)
        tdm_load_2d((unsigned)(size_t)(void*)dst, src, 4096u, 1u, 4096u);
#else
    for (int it = 0; it < 16; ++it) {
        const int idx = it * 256 + (int)threadIdx.x;
        dst[idx] = src[idx];
    }
#endif
}

// Wave0 drains TENSORcnt (no barrier).
__device__ __forceinline__ void tdm_wait_nobar() {
#if USE_TDM
    if (threadIdx.x < 32) __builtin_amdgcn_s_wait_tensorcnt(0);
#endif
}

// Drain TENSORcnt then publish LDS to the whole workgroup.
__device__ __forceinline__ void tile_wait() {
    tdm_wait_nobar();
    __syncthreads();
}

// Workgroup (256 threads = 8 wave32) 64x64 fp32 matmul entirely in LDS.
// C = A * B, row-major stride 64. Each wave computes 2 of 16 16x16 tiles,
// K swept in steps of 4 with v_wmma_f32_16x16x4_f32 (ISA 7.12.2 frag layout).
__device__ void wg_mm64(float* __restrict__ C, const float* __restrict__ A,
                        const float* __restrict__ B) {
    __syncthreads();
    const int tid  = threadIdx.x;
    const int wave = tid >> 5;
    const int lane = tid & 31;
    const int mrow = lane & 15;
    const int khh  = lane >> 4;      // 0 or 1
    const int kh   = khh << 1;       // 0 or 2
    for (int t = wave; t < 16; t += 8) {
        const int ti = (t >> 2) << 4;
        const int tj = (t & 3) << 4;
        v8f acc = {};
        for (int kb = 0; kb < 64; kb += 4) {
            v2f a, b;
            a.x = A[(ti + mrow) * 64 + kb + kh];
            a.y = A[(ti + mrow) * 64 + kb + kh + 1];
            b.x = B[(kb + kh) * 64 + tj + mrow];
            b.y = B[(kb + kh + 1) * 64 + tj + mrow];
            acc = wmma4(a, b, acc);
        }
        const int rbase = ti + (khh << 3);
#pragma unroll
        for (int r = 0; r < 8; ++r)
            C[(rbase + r) * 64 + tj + mrow] = acc[r];
    }
    __syncthreads();
}

// ---------------------------------------------------------------------------
// Kernel 1: per-head E = expm(X_h) (scaling-and-squaring j=6, Horner order 12)
// then binary power table Epow[h][k] = E^(2^k), k=0..9.  grid=16, 256 thr.
// ---------------------------------------------------------------------------
__global__ void expm_powers(const float* __restrict__ X, float* __restrict__ Epow) {
    __shared__ float As[4096];
    __shared__ float Pbuf[4096];
    __shared__ float Tbuf[4096];
    const int h = blockIdx.x;
    const int tid = threadIdx.x;
    const float inv_scale = 1.0f / 64.0f;   // 2^-6
    for (int idx = tid; idx < 4096; idx += 256)
        As[idx] = X[h * 4096 + idx] * inv_scale;
    for (int idx = tid; idx < 4096; idx += 256)
        Pbuf[idx] = ((idx >> 6) == (idx & 63)) ? 1.0f : 0.0f;
    for (int k = 12; k >= 1; --k) {
        wg_mm64(Tbuf, As, Pbuf);
        const float invk = 1.0f / (float)k;
        for (int idx = tid; idx < 4096; idx += 256)
            Pbuf[idx] = (((idx >> 6) == (idx & 63)) ? 1.0f : 0.0f) + Tbuf[idx] * invk;
        __syncthreads();
    }
    float* P = Pbuf;
    float* T = Tbuf;
    for (int j = 0; j < 6; ++j) {
        wg_mm64(T, P, P);
        float* tmp = P; P = T; T = tmp;
    }
    for (int k = 0; k < 10; ++k) {
        for (int idx = tid; idx < 4096; idx += 256)
            Epow[(h * 10 + k) * 4096 + idx] = P[idx];
        if (k < 9) {
            wg_mm64(T, P, P);
            float* tmp = P; P = T; T = tmp;
        }
    }
}

// ---------------------------------------------------------------------------
// Kernel 2: GEMM Out = Xin(2048x1024) * W^T + bias, TDM double-buffered.
// mode 0: scatter to (B,H,L,64) (QKV). mode 1: flat (B,L,DIM).
// grid (16,16): 128x64 tile/WG. LDS panels: A[128x16] row-major, B[64x16]
// as [n][k] so the TDM copy needs no transpose (WMMA B-frag = Bl[n*16+k]).
// ---------------------------------------------------------------------------
__device__ __forceinline__ void gemm_panels_issue(float* Adst, float* Bdst,
                                                  const float* Xin, const float* W,
                                                  int m0, int n0, int kbase) {
#if USE_TDM
    if (threadIdx.x < 32) {
        tdm_load_2d((unsigned)(size_t)(void*)Adst,
                    Xin + (size_t)m0 * 1024 + kbase, 16u, 128u, 1024u);
        tdm_load_2d((unsigned)(size_t)(void*)Bdst,
                    W + (size_t)n0 * 1024 + kbase, 16u, 64u, 1024u);
    }
#else
    const int lr = (int)threadIdx.x >> 4;
    const int lc = (int)threadIdx.x & 15;
#pragma unroll
    for (int it = 0; it < 8; ++it) {
        const int row = it * 16 + lr;
        Adst[row * 16 + lc] = Xin[(size_t)(m0 + row) * 1024 + kbase + lc];
    }
#pragma unroll
    for (int it = 0; it < 4; ++it) {
        const int n = it * 16 + lr;
        Bdst[n * 16 + lc] = W[(size_t)(n0 + n) * 1024 + kbase + lc];
    }
#endif
}

__global__ void gemm2048(const float* __restrict__ Xin, const float* __restrict__ W,
                         const float* __restrict__ bias, float* __restrict__ Out,
                         int mode) {
    __shared__ float Al[2][128 * 16];
    __shared__ float Bl[2][64 * 16];
    const int m0 = blockIdx.x * 128;
    const int n0 = blockIdx.y * 64;
    const int tid  = threadIdx.x;
    const int wave = tid >> 5;
    const int lane = tid & 31;
    const int mrow = lane & 15;
    const int khh  = lane >> 4;
    const int kh   = khh << 1;
    v8f acc[4] = {};

    gemm_panels_issue(Al[0], Bl[0], Xin, W, m0, n0, 0);
    tile_wait();
    for (int kt = 0; kt < 64; ++kt) {
        const int cur = kt & 1;
        const int nxt = cur ^ 1;
        if (kt + 1 < 64)
            gemm_panels_issue(Al[nxt], Bl[nxt], Xin, W, m0, n0, (kt + 1) * 16);
        const float* Ap = Al[cur];
        const float* Bp = Bl[cur];
#pragma unroll
        for (int kb = 0; kb < 16; kb += 4) {
            v2f a;
            a.x = Ap[(wave * 16 + mrow) * 16 + kb + kh];
            a.y = Ap[(wave * 16 + mrow) * 16 + kb + kh + 1];
#pragma unroll
            for (int tj = 0; tj < 4; ++tj) {
                v2f b;   // B[k][n] = W[n][k] = Bp[n*16 + k]
                b.x = Bp[(tj * 16 + mrow) * 16 + kb + kh];
                b.y = Bp[(tj * 16 + mrow) * 16 + kb + kh + 1];
                acc[tj] = wmma4(a, b, acc[tj]);
            }
        }
        tile_wait();   // next panels ready + everyone done with cur
    }
#pragma unroll
    for (int tj = 0; tj < 4; ++tj) {
#pragma unroll
        for (int r = 0; r < 8; ++r) {
            const int m = m0 + wave * 16 + r + (khh << 3);
            const int n = n0 + tj * 16 + mrow;
            const float val = acc[tj][r] + bias[n];
            if (mode == 0) {
                const int b = m >> 10, l = m & 1023, hh = n >> 6, dd = n & 63;
                Out[(((size_t)(b * 16 + hh)) * 1024 + l) * 64 + dd] = val;
            } else {
                Out[(size_t)m * 1024 + n] = val;
            }
        }
    }
}

// ---------------------------------------------------------------------------
// Kernel 3: Q_ref/K_ref transform. Tinv(l) = sqrt(s_h) * (R_h^l)^T (R
// orthogonal). Per WG: one head, one 64-position chunk. R^{l0} rebuilt from
// the power table (TDM-loaded), then R stepped by E per position via WMMA.
// grid = 256 (h*16+chunk), 256 threads.
// ---------------------------------------------------------------------------
__global__ void transform_qk(const float* __restrict__ Q, const float* __restrict__ K,
                             const float* __restrict__ Epow, const float* __restrict__ log_s,
                             float* __restrict__ Qr, float* __restrict__ Kr) {
    __shared__ float bufA[4096];
    __shared__ float bufB[4096];
    __shared__ float Eb[4096];
    __shared__ float vbuf[256];
    const int h  = blockIdx.x >> 4;
    const int l0 = (blockIdx.x & 15) * 64;
    const int tid = threadIdx.x;
    const float sc = expf(0.5f * log_s[h]);
    float* Rc = bufA;
    float* Rt = bufB;
    for (int idx = tid; idx < 4096; idx += 256)
        Rc[idx] = ((idx >> 6) == (idx & 63)) ? 1.0f : 0.0f;
    for (int k = 6; k <= 9; ++k) {
        if ((l0 >> k) & 1) {   // uniform per-WG
            tile16k_issue(Eb, Epow + (size_t)(h * 10 + k) * 4096);
            tdm_wait_nobar();
            wg_mm64(Rt, Rc, Eb);   // entry barrier publishes Eb
            float* tmp = Rc; Rc = Rt; Rt = tmp;
        }
    }
    tile16k_issue(Eb, Epow + (size_t)(h * 10 + 0) * 4096);
    tdm_wait_nobar();
    const int vec  = tid >> 6;   // 0: Q b0, 1: Q b1, 2: K b0, 3: K b1
    const int comp = tid & 63;
    const int b    = vec & 1;
    const float* src = (vec & 2) ? K : Q;
    float*       dst = (vec & 2) ? Kr : Qr;
    const size_t vbase = ((size_t)(b * 16 + h)) * 1024 * 64;
    for (int p = 0; p < 64; ++p) {
        const int l = l0 + p;
        __syncthreads();
        vbuf[tid] = src[vbase + (size_t)l * 64 + comp];
        __syncthreads();
        // out_i = sc * sum_j R[j][i] * v[j]   (transpose-apply of R^l)
        float o = 0.0f;
        const float* vv = &vbuf[vec * 64];
        for (int j = 0; j < 64; ++j)
            o += Rc[j * 64 + comp] * vv[j];
        dst[vbase + (size_t)l * 64 + comp] = o * sc;
        if (p != 63) {
            wg_mm64(Rt, Rc, Eb);   // R^{l+1} = R^l * E
            float* tmp = Rc; Rc = Rt; Rt = tmp;
        }
    }
}

// ---------------------------------------------------------------------------
// Kernel 4: flash attention with Gaussian locality log-bias.
// grid (16 iblocks, 16 heads, 2 batch). 64 query rows/WG, streamed over 16
// key tiles with online softmax; K/V/Q tiles DMA'd by the TDM. The score
// tile reuses the K-tile LDS buffer after a barrier.
// ---------------------------------------------------------------------------
__global__ void flash_attn(const float* __restrict__ Qr, const float* __restrict__ Kr,
                           const float* __restrict__ V, const int* __restrict__ positions,
                           const float* __restrict__ log_sigma, float* __restrict__ AO) {
    __shared__ float Qi[4096];
    __shared__ float KS[4096];   // K tile, then (after barrier) score/P tile
    __shared__ float Vl[4096];
    __shared__ float rowm[64];
    __shared__ float rowl[64];
    __shared__ float rowc[64];
    const int i0 = blockIdx.x * 64;
    const int h  = blockIdx.y;
    const int b  = blockIdx.z;
    const int tid  = threadIdx.x;
    const int wave = tid >> 5;
    const int lane = tid & 31;
    const int mrow = lane & 15;
    const int khh  = lane >> 4;
    const int kh   = khh << 1;
    const float sigma  = expf(log_sigma[0]);
    const float inv2s2 = 1.0f / (2.0f * sigma * sigma);
    const size_t base = ((size_t)(b * 16 + h)) * 1024 * 64;

    tile16k_issue(Qi, Qr + base + (size_t)i0 * 64);
    if (tid < 64) { rowm[tid] = -1e30f; rowl[tid] = 0.0f; }
    v8f oacc[2] = {};
    tile_wait();
    for (int jb = 0; jb < 16; ++jb) {
        const int j0 = jb * 64;
        tile16k_issue(KS, Kr + base + (size_t)j0 * 64);
        tile16k_issue(Vl, V + base + (size_t)j0 * 64);
        tile_wait();
        // S = Qi * Kj^T  (B[k][n] = K[n][k]: read K rows per lane)
        v8f sacc[2];
#pragma unroll
        for (int tt = 0; tt < 2; ++tt) {
            const int t  = wave + tt * 8;
            const int ti = (t >> 2) << 4;
            const int tj = (t & 3) << 4;
            v8f acc = {};
            for (int kb = 0; kb < 64; kb += 4) {
                v2f a, bf;
                a.x  = Qi[(ti + mrow) * 64 + kb + kh];
                a.y  = Qi[(ti + mrow) * 64 + kb + kh + 1];
                bf.x = KS[(tj + mrow) * 64 + kb + kh];
                bf.y = KS[(tj + mrow) * 64 + kb + kh + 1];
                acc = wmma4(a, bf, acc);
            }
            sacc[tt] = acc;
        }
        __syncthreads();   // all waves done reading K tile
        // scaled scores + locality log-bias into KS
#pragma unroll
        for (int tt = 0; tt < 2; ++tt) {
            const int t  = wave + tt * 8;
            const int ti = (t >> 2) << 4;
            const int tj = (t & 3) << 4;
#pragma unroll
            for (int r = 0; r < 8; ++r) {
                const int ig = i0 + ti + r + (khh << 3);
                const int jg = j0 + tj + mrow;
                const float d = (float)(positions[ig] - positions[jg]);
                const float bias = logf(expf(-d * d * inv2s2) + 1e-6f);
                KS[(ti + r + (khh << 3)) * 64 + tj + mrow] = sacc[tt][r] * 0.125f + bias;
            }
        }
        __syncthreads();
        // online softmax row update (one thread per query row)
        if (tid < 64) {
            float mx = rowm[tid];
            for (int j = 0; j < 64; ++j) mx = fmaxf(mx, KS[tid * 64 + j]);
            const float c = expf(rowm[tid] - mx);
            float s = rowl[tid] * c;
            for (int j = 0; j < 64; ++j) {
                const float p = expf(KS[tid * 64 + j] - mx);
                KS[tid * 64 + j] = p;
                s += p;
            }
            rowm[tid] = mx; rowl[tid] = s; rowc[tid] = c;
        }
        __syncthreads();
        // O = O*corr + P * Vj
#pragma unroll
        for (int tt = 0; tt < 2; ++tt) {
            const int t  = wave + tt * 8;
            const int ti = (t >> 2) << 4;
            const int tj = (t & 3) << 4;
            v8f acc = oacc[tt];
#pragma unroll
            for (int r = 0; r < 8; ++r)
                acc[r] *= rowc[ti + r + (khh << 3)];
            for (int kb = 0; kb < 64; kb += 4) {
                v2f a, bf;
                a.x  = KS[(ti + mrow) * 64 + kb + kh];
                a.y  = KS[(ti + mrow) * 64 + kb + kh + 1];
                bf.x = Vl[(kb + kh) * 64 + tj + mrow];
                bf.y = Vl[(kb + kh + 1) * 64 + tj + mrow];
                acc = wmma4(a, bf, acc);
            }
            oacc[tt] = acc;
        }
        __syncthreads();
    }
    // finalize and scatter to (B, L, DIM) layout
#pragma unroll
    for (int tt = 0; tt < 2; ++tt) {
        const int t  = wave + tt * 8;
        const int ti = (t >> 2) << 4;
        const int tj = (t & 3) << 4;
#pragma unroll
        for (int r = 0; r < 8; ++r) {
            const int row = ti + r + (khh << 3);
            const float val = oacc[tt][r] / rowl[row];
            AO[((size_t)(b * 1024) + i0 + row) * 1024 + h * 64 + tj + mrow] = val;
        }
    }
}

// ---------------------------------------------------------------------------
extern "C" void kernel_launch(void* const* d_in, const int* in_sizes, int n_in,
                              void* d_out, int out_size, void* d_ws, size_t ws_size,
                              hipStream_t stream) {
    (void)in_sizes; (void)n_in; (void)out_size; (void)ws_size;
    const float* x         = (const float*)d_in[0];
    const int*   positions = (const int*)d_in[1];
    const float* Wq = (const float*)d_in[2];
    const float* bq = (const float*)d_in[3];
    const float* Wk = (const float*)d_in[4];
    const float* bk = (const float*)d_in[5];
    const float* Wv = (const float*)d_in[6];
    const float* bv = (const float*)d_in[7];
    const float* Wo = (const float*)d_in[8];
    const float* bo = (const float*)d_in[9];
    const float* log_s     = (const float*)d_in[10];
    const float* X         = (const float*)d_in[11];
    const float* log_sigma = (const float*)d_in[12];
    float* out = (float*)d_out;
    float* ws  = (float*)d_ws;

    const size_t NQ = (size_t)2 * 16 * 1024 * 64;   // 2,097,152 floats
    float* Q    = ws;
    float* K    = Q  + NQ;
    float* V    = K  + NQ;
    float* Qr   = V  + NQ;
    float* Kr   = Qr + NQ;
    float* AO   = Kr + NQ;
    float* Epow = AO + (size_t)2 * 1024 * 1024;      // 16*10*4096 floats

    expm_powers<<<16, 256, 0, stream>>>(X, Epow);
    gemm2048<<<dim3(16, 16), 256, 0, stream>>>(x, Wq, bq, Q, 0);
    gemm2048<<<dim3(16, 16), 256, 0, stream>>>(x, Wk, bk, K, 0);
    gemm2048<<<dim3(16, 16), 256, 0, stream>>>(x, Wv, bv, V, 0);
    transform_qk<<<256, 256, 0, stream>>>(Q, K, Epow, log_s, Qr, Kr);
    flash_attn<<<dim3(16, 16, 2), 256, 0, stream>>>(Qr, Kr, V, positions, log_sigma, AO);
    gemm2048<<<dim3(16, 16), 256, 0, stream>>>(AO, Wo, bo, out, 1);
}